// GraphSAGEModel_19473381720256
// MI455X (gfx1250) — compile-verified
//
#include <hip/hip_runtime.h>
#include <hip/hip_bf16.h>

// GraphSAGE 3-layer forward for MI455X (gfx1250, wave32).
// GEMMs use V_WMMA_F32_16X16X4_F32 (full fp32 precision, matches f32 reference).
// Aggregation uses float4 gathers + global_atomic_add_f32 (L2-resident working set).

typedef __attribute__((ext_vector_type(2))) float v2f;
typedef __attribute__((ext_vector_type(8))) float v8f;

#define DIM 128            // input/hidden feature width (all layers' K)
#define BN_EPS 1e-5f

// ---------------------------------------------------------------- utilities
__global__ void zero_kernel(float* __restrict__ p, int n) {
  int i = blockIdx.x * blockDim.x + threadIdx.x;
  if (i < n) p[i] = 0.0f;
}

// ------------------------------------------------------------- degree count
__global__ void count_edges_kernel(const long long* __restrict__ ei,
                                   float* __restrict__ cnt, int nEdges) {
  int e = blockIdx.x * blockDim.x + threadIdx.x;
  if (e < nEdges) {
    long long d = ei[nEdges + e];           // dst row of edge_index
    atomicAdd(cnt + d, 1.0f);
  }
}

// -------------------------------------------------- scatter-add aggregation
// One thread handles 4 contiguous features of one edge: a wave (32 lanes)
// covers one full 128-feature row -> fully coalesced 512B gather per edge.
__global__ void scatter_add4_kernel(const long long* __restrict__ ei,
                                    const float* __restrict__ feat,
                                    float* __restrict__ agg, int nEdges) {
  int gid = blockIdx.x * blockDim.x + threadIdx.x;
  int total = nEdges * (DIM / 4);
  if (gid >= total) return;
  int e = gid >> 5;                          // 32 quads per edge
  int f = (gid & 31) << 2;                   // feature start
  long long s = ei[e];
  long long d = ei[nEdges + e];
  const float4 v = *(const float4*)(feat + (size_t)s * DIM + f);
  float* dst = agg + (size_t)d * DIM + f;
  atomicAdd(dst + 0, v.x);
  atomicAdd(dst + 1, v.y);
  atomicAdd(dst + 2, v.z);
  atomicAdd(dst + 3, v.w);
}

// agg[i][f] /= max(cnt[i], 1)
__global__ void mean_kernel(float* __restrict__ agg,
                            const float* __restrict__ cnt, int nNodes) {
  int gid = blockIdx.x * blockDim.x + threadIdx.x;
  if (gid >= nNodes * DIM) return;
  float c = cnt[gid >> 7];                   // DIM == 128
  agg[gid] = agg[gid] / fmaxf(c, 1.0f);
}

// ------------------------------------------------------- dual-GEMM (WMMA)
// out[row][col] = sum_k A1[row][k]*W1[col][k] + sum_k A2[row][k]*W2[col][k] + bias[col]
// One wave computes one 16x16 output tile with V_WMMA_F32_16X16X4_F32.
// f32 A-frag (16x4): lane L (hi=L>>4, m=L&15) holds A[m][k0+2hi], A[m][k0+2hi+1].
// f32 B-frag (4x16): lane L holds B[k0+2hi][n=L&15], B[k0+2hi+1][n]; B[k][n]=W[n][k]
//   -> both frags are single 8-byte loads from row-major storage.
// C/D (8 VGPRs): acc[r] -> row r + 8*hi, col L&15.
__global__ void gemm_dual_wmma(const float* __restrict__ A1, const float* __restrict__ W1,
                               const float* __restrict__ A2, const float* __restrict__ W2,
                               const float* __restrict__ bias, float* __restrict__ out,
                               int nRows, int OUT) {
  const int lane = threadIdx.x & 31;
  const int wid  = blockIdx.x * (blockDim.x >> 5) + (threadIdx.x >> 5);
  const int nTilesCol = OUT >> 4;
  const int tileM = wid / nTilesCol;
  const int tileN = wid % nTilesCol;
  if (tileM * 16 >= nRows) return;           // wave-uniform guard (EXEC stays all-1s)

  const int hi  = lane >> 4;                 // 0 or 1
  const int l15 = lane & 15;
  const size_t rowA = (size_t)(tileM * 16 + l15) * DIM;   // A row for this lane
  const size_t rowW = (size_t)(tileN * 16 + l15) * DIM;   // W row (== output col)

  v8f acc = {};
#pragma unroll
  for (int k0 = 0; k0 < DIM; k0 += 4) {
    const int ka = k0 + 2 * hi;
    v2f a1 = *(const v2f*)(A1 + rowA + ka);
    v2f b1 = *(const v2f*)(W1 + rowW + ka);
    acc = __builtin_amdgcn_wmma_f32_16x16x4_f32(false, a1, false, b1,
                                                (short)0, acc, false, false);
    v2f a2 = *(const v2f*)(A2 + rowA + ka);
    v2f b2 = *(const v2f*)(W2 + rowW + ka);
    acc = __builtin_amdgcn_wmma_f32_16x16x4_f32(false, a2, false, b2,
                                                (short)0, acc, false, false);
  }

  const float bv = bias[tileN * 16 + l15];
#pragma unroll
  for (int r = 0; r < 8; ++r) {
    int row = tileM * 16 + r + 8 * hi;
    out[(size_t)row * OUT + tileN * 16 + l15] = acc[r] + bv;
  }
}

// ---------------------------------------------------------------- BatchNorm
// Hierarchical stats: each block (128 threads = 1 thread/feature) accumulates
// sum & sumsq over `rowsPerBlock` rows in registers, then 2 atomics/feature.
__global__ void bn_stats_kernel(const float* __restrict__ h,
                                float* __restrict__ stats,
                                int nRows, int rowsPerBlock) {
  int f  = threadIdx.x;                      // 0..127
  int r0 = blockIdx.x * rowsPerBlock;
  int r1 = min(r0 + rowsPerBlock, nRows);
  float s = 0.0f, sq = 0.0f;
  for (int r = r0; r < r1; ++r) {
    float v = h[(size_t)r * DIM + f];
    s  += v;
    sq += v * v;
  }
  atomicAdd(&stats[f],        s);
  atomicAdd(&stats[DIM + f],  sq);
}

__global__ void bn_relu_kernel(float* __restrict__ h,
                               const float* __restrict__ stats,
                               const float* __restrict__ g,
                               const float* __restrict__ be,
                               int nNodes, float invN) {
  int gid = blockIdx.x * blockDim.x + threadIdx.x;
  if (gid >= nNodes * DIM) return;
  int f = gid & (DIM - 1);
  float mu  = stats[f] * invN;
  float var = stats[DIM + f] * invN - mu * mu;
  float inv = 1.0f / sqrtf(var + BN_EPS);
  float v = (h[gid] - mu) * inv * g[f] + be[f];
  h[gid] = v > 0.0f ? v : 0.0f;
}

// ---------------------------------------------------------------- launcher
extern "C" void kernel_launch(void* const* d_in, const int* in_sizes, int n_in,
                              void* d_out, int out_size, void* d_ws, size_t ws_size,
                              hipStream_t stream) {
  const float*     x   = (const float*)d_in[0];
  const long long* ei  = (const long long*)d_in[1];   // int64 edge_index [2, E]
  const float* Wl1 = (const float*)d_in[2];
  const float* Wr1 = (const float*)d_in[3];
  const float* b1  = (const float*)d_in[4];
  const float* g1  = (const float*)d_in[5];
  const float* be1 = (const float*)d_in[6];
  const float* Wl2 = (const float*)d_in[7];
  const float* Wr2 = (const float*)d_in[8];
  const float* b2  = (const float*)d_in[9];
  const float* g2  = (const float*)d_in[10];
  const float* be2 = (const float*)d_in[11];
  const float* Wl3 = (const float*)d_in[12];
  const float* Wr3 = (const float*)d_in[13];
  const float* b3  = (const float*)d_in[14];
  float* out = (float*)d_out;

  const int nNodes = in_sizes[0] / DIM;      // 50000
  const int nEdges = in_sizes[1] / 2;        // 800000
  const int outDim = in_sizes[12] / DIM;     // 64
  const float invN = 1.0f / (float)nNodes;

  // workspace layout (floats): cnt | agg | h1 | h2 | stats   (~77 MB)
  float* ws = (float*)d_ws;
  const size_t nodeFeat = (size_t)nNodes * DIM;
  float* cnt   = ws;
  float* agg   = cnt + (((size_t)nNodes + 255) & ~(size_t)255);
  float* h1    = agg + nodeFeat;
  float* h2    = h1 + nodeFeat;
  float* stats = h2 + nodeFeat;

  const int TB = 256;
  const int nNF = nNodes * DIM;
  const int nScatter = nEdges * (DIM / 4);
  const int gNF = (nNF + TB - 1) / TB;
  const int gSc = (nScatter + TB - 1) / TB;
  const int statBlocks = (nNodes + 255) / 256;

  // in-degree counts (shared by all layers)
  zero_kernel<<<(nNodes + TB - 1) / TB, TB, 0, stream>>>(cnt, nNodes);
  count_edges_kernel<<<(nEdges + TB - 1) / TB, TB, 0, stream>>>(ei, cnt, nEdges);

  // ---- layer 1: h1 = mean(x) @ Wl1^T + x @ Wr1^T + b1 ; BN ; ReLU
  zero_kernel<<<gNF, TB, 0, stream>>>(agg, nNF);
  scatter_add4_kernel<<<gSc, TB, 0, stream>>>(ei, x, agg, nEdges);
  mean_kernel<<<gNF, TB, 0, stream>>>(agg, cnt, nNodes);
  {
    int waves = (nNodes / 16) * (DIM / 16);
    gemm_dual_wmma<<<(waves + 3) / 4, 128, 0, stream>>>(agg, Wl1, x, Wr1, b1, h1, nNodes, DIM);
  }
  zero_kernel<<<1, 256, 0, stream>>>(stats, 2 * DIM);
  bn_stats_kernel<<<statBlocks, DIM, 0, stream>>>(h1, stats, nNodes, 256);
  bn_relu_kernel<<<gNF, TB, 0, stream>>>(h1, stats, g1, be1, nNodes, invN);

  // ---- layer 2: h2 = mean(h1) @ Wl2^T + h1 @ Wr2^T + b2 ; BN ; ReLU
  zero_kernel<<<gNF, TB, 0, stream>>>(agg, nNF);
  scatter_add4_kernel<<<gSc, TB, 0, stream>>>(ei, h1, agg, nEdges);
  mean_kernel<<<gNF, TB, 0, stream>>>(agg, cnt, nNodes);
  {
    int waves = (nNodes / 16) * (DIM / 16);
    gemm_dual_wmma<<<(waves + 3) / 4, 128, 0, stream>>>(agg, Wl2, h1, Wr2, b2, h2, nNodes, DIM);
  }
  zero_kernel<<<1, 256, 0, stream>>>(stats, 2 * DIM);
  bn_stats_kernel<<<statBlocks, DIM, 0, stream>>>(h2, stats, nNodes, 256);
  bn_relu_kernel<<<gNF, TB, 0, stream>>>(h2, stats, g2, be2, nNodes, invN);

  // ---- layer 3: out = mean(h2) @ Wl3^T + h2 @ Wr3^T + b3
  zero_kernel<<<gNF, TB, 0, stream>>>(agg, nNF);
  scatter_add4_kernel<<<gSc, TB, 0, stream>>>(ei, h2, agg, nEdges);
  mean_kernel<<<gNF, TB, 0, stream>>>(agg, cnt, nNodes);
  {
    int waves = (nNodes / 16) * (outDim / 16);
    gemm_dual_wmma<<<(waves + 3) / 4, 128, 0, stream>>>(agg, Wl3, h2, Wr3, b3, out, nNodes, outDim);
  }
}